// SuperpixelColor_66417374265608
// MI455X (gfx1250) — compile-verified
//
#include <hip/hip_runtime.h>
#include <cstdint>

// ---------------------------------------------------------------------------
// Superpixel color averaging, MI455X (gfx1250).
// B=16, C=3, H=W=1024, n_sp = 32*32 = 1024.
// Memory-bound (~550 MB total traffic -> ~24 us at 23.3 TB/s). No matmul
// structure -> WMMA not applicable; we use the CDNA5 async global->LDS path,
// wave32 LDS float atomics, and b128 vector memory ops instead.
// ---------------------------------------------------------------------------

#define NIMG 16
#define NSP 1024            // superpixels per image
#define PIX (1024 * 1024)   // pixels per image
#define TPB 256             // 8 waves of 32 on CDNA5
#define ABLK 128            // accumulation blocks per image (8192 px each)
#define GBLK 128            // gather blocks per image (8192 px each)

// 16-byte vector type matching the async-LDS builtin's pointee type.
typedef int v4i __attribute__((vector_size(16)));
typedef __attribute__((address_space(1))) v4i* gptr_v4i;
typedef __attribute__((address_space(3))) v4i* lptr_v4i;

// ---- CDNA5 async global->LDS helpers --------------------------------------
#if defined(__has_builtin)
#if __has_builtin(__builtin_amdgcn_global_load_async_to_lds_b128)
#define HAVE_ASYNC_BUILTIN 1
#else
#define HAVE_ASYNC_BUILTIN 0
#endif
#if __has_builtin(__builtin_amdgcn_s_wait_asynccnt)
#define HAVE_WAIT_ASYNC_BUILTIN 1
#else
#define HAVE_WAIT_ASYNC_BUILTIN 0
#endif
#else
#define HAVE_ASYNC_BUILTIN 0
#define HAVE_WAIT_ASYNC_BUILTIN 0
#endif

__device__ __forceinline__ void async_copy16_g2l(const void* gsrc, void* lds_dst) {
#if HAVE_ASYNC_BUILTIN
  __builtin_amdgcn_global_load_async_to_lds_b128(
      (gptr_v4i)(uintptr_t)gsrc,
      (lptr_v4i)(uint32_t)(uintptr_t)lds_dst,
      0, 0);
#else
  // LDS dest byte-offset lives in a 32-bit VGPR; 64-bit global addr in a pair.
  uint32_t l = (uint32_t)(uintptr_t)lds_dst;
  uint64_t g = (uint64_t)(uintptr_t)gsrc;
  asm volatile("global_load_async_to_lds_b128 %0, %1, off"
               :
               : "v"(l), "v"(g)
               : "memory");
#endif
}

__device__ __forceinline__ void wait_async_zero() {
#if HAVE_WAIT_ASYNC_BUILTIN
  __builtin_amdgcn_s_wait_asynccnt(0);
#else
  asm volatile("s_wait_asynccnt 0" ::: "memory");
#endif
}

// ---------------------------------------------------------------------------
// Kernel 1: zero the workspace (B * NSP float4 of {sumR,sumG,sumB,count}).
// Must run every launch: we atomically accumulate into it and the harness
// does not re-poison between replays.
// ---------------------------------------------------------------------------
__global__ void sp_zero_ws(float4* __restrict__ ws4) {
  int i = blockIdx.x * blockDim.x + threadIdx.x;
  if (i < NIMG * NSP) ws4[i] = make_float4(0.f, 0.f, 0.f, 0.f);
}

// ---------------------------------------------------------------------------
// Kernel 2: per-block LDS-privatized segment accumulation, then atomic flush.
// Each block: one image, 8192 pixels. LDS table = 1024 x float4 = 16 KB.
// ---------------------------------------------------------------------------
__global__ void __launch_bounds__(TPB)
sp_accumulate(const float* __restrict__ in, const int* __restrict__ lbl,
              float* __restrict__ ws) {
  __shared__ float4 acc[NSP];  // {r,g,b,count} per superpixel

  for (int i = threadIdx.x; i < NSP; i += TPB)
    acc[i] = make_float4(0.f, 0.f, 0.f, 0.f);
  __syncthreads();

  const int b = blockIdx.x / ABLK;
  const int blk = blockIdx.x % ABLK;
  const int chunk = PIX / ABLK;  // 8192 pixels
  const size_t poff = (size_t)blk * chunk;

  const int4* __restrict__ l4 =
      (const int4*)(lbl + (size_t)b * PIX + poff);
  const float4* __restrict__ r4 =
      (const float4*)(in + (size_t)b * 3 * PIX + 0 * (size_t)PIX + poff);
  const float4* __restrict__ g4 =
      (const float4*)(in + (size_t)b * 3 * PIX + 1 * (size_t)PIX + poff);
  const float4* __restrict__ b4 =
      (const float4*)(in + (size_t)b * 3 * PIX + 2 * (size_t)PIX + poff);

  const int n4 = chunk / 4;  // 2048 int4/float4 groups
  for (int i = threadIdx.x; i < n4; i += TPB) {
    // cover latency of the upcoming strided pass (speculative, OOB-safe)
    __builtin_prefetch(&l4[i + TPB], 0, 0);
    __builtin_prefetch(&r4[i + TPB], 0, 0);
    __builtin_prefetch(&g4[i + TPB], 0, 0);
    __builtin_prefetch(&b4[i + TPB], 0, 0);

    const int4 L = l4[i];
    const float4 R = r4[i], G = g4[i], Bv = b4[i];

    float* a0 = (float*)&acc[L.x];
    atomicAdd(a0 + 0, R.x); atomicAdd(a0 + 1, G.x);
    atomicAdd(a0 + 2, Bv.x); atomicAdd(a0 + 3, 1.0f);
    float* a1 = (float*)&acc[L.y];
    atomicAdd(a1 + 0, R.y); atomicAdd(a1 + 1, G.y);
    atomicAdd(a1 + 2, Bv.y); atomicAdd(a1 + 3, 1.0f);
    float* a2 = (float*)&acc[L.z];
    atomicAdd(a2 + 0, R.z); atomicAdd(a2 + 1, G.z);
    atomicAdd(a2 + 2, Bv.z); atomicAdd(a2 + 3, 1.0f);
    float* a3 = (float*)&acc[L.w];
    atomicAdd(a3 + 0, R.w); atomicAdd(a3 + 1, G.w);
    atomicAdd(a3 + 2, Bv.w); atomicAdd(a3 + 3, 1.0f);
  }
  __syncthreads();

  // Flush block-local partials to the per-image global table.
  float* __restrict__ gdst = ws + (size_t)b * NSP * 4;
  const float* __restrict__ asrc = (const float*)acc;
  for (int i = threadIdx.x; i < NSP * 4; i += TPB) {
    const float v = asrc[i];
    if (v != 0.0f) atomicAdd(gdst + i, v);  // global_atomic_add_f32
  }
}

// ---------------------------------------------------------------------------
// Kernel 3: async-load the 16 KB sums table into LDS, convert to means in
// place, then gather means back to every pixel (3 channel-planar streams).
// ---------------------------------------------------------------------------
__global__ void __launch_bounds__(TPB)
sp_gather(const int* __restrict__ lbl, const float* __restrict__ ws,
          float* __restrict__ out) {
  __shared__ float4 means[NSP];  // 16 KB

  const int b = blockIdx.x / GBLK;
  const int blk = blockIdx.x % GBLK;

  // CDNA5 async global->LDS broadcast of this image's sums table.
  const float4* __restrict__ sums = (const float4*)(ws + (size_t)b * NSP * 4);
  for (int i = threadIdx.x; i < NSP; i += TPB)
    async_copy16_g2l(&sums[i], &means[i]);
  wait_async_zero();   // my waves' transfers landed
  __syncthreads();     // all waves' transfers landed

  // sums -> means in place.
  for (int i = threadIdx.x; i < NSP; i += TPB) {
    const float4 s = means[i];
    const float inv = 1.0f / fmaxf(s.w, 1.0f);
    means[i] = make_float4(s.x * inv, s.y * inv, s.z * inv, s.w);
  }
  __syncthreads();

  const int chunk = PIX / GBLK;  // 8192 pixels
  const size_t poff = (size_t)blk * chunk;
  const int4* __restrict__ l4 = (const int4*)(lbl + (size_t)b * PIX + poff);
  float4* __restrict__ outr =
      (float4*)(out + (size_t)b * 3 * PIX + 0 * (size_t)PIX + poff);
  float4* __restrict__ outg =
      (float4*)(out + (size_t)b * 3 * PIX + 1 * (size_t)PIX + poff);
  float4* __restrict__ outb =
      (float4*)(out + (size_t)b * 3 * PIX + 2 * (size_t)PIX + poff);

  const int n4 = chunk / 4;
  for (int i = threadIdx.x; i < n4; i += TPB) {
    __builtin_prefetch(&l4[i + TPB], 0, 0);
    const int4 L = l4[i];
    const float4 m0 = means[L.x];
    const float4 m1 = means[L.y];
    const float4 m2 = means[L.z];
    const float4 m3 = means[L.w];
    outr[i] = make_float4(m0.x, m1.x, m2.x, m3.x);
    outg[i] = make_float4(m0.y, m1.y, m2.y, m3.y);
    outb[i] = make_float4(m0.z, m1.z, m2.z, m3.z);
  }
}

// ---------------------------------------------------------------------------
extern "C" void kernel_launch(void* const* d_in, const int* in_sizes, int n_in,
                              void* d_out, int out_size, void* d_ws,
                              size_t ws_size, hipStream_t stream) {
  (void)in_sizes; (void)n_in; (void)out_size; (void)ws_size;
  const float* input = (const float*)d_in[0];     // [16,3,1024,1024] f32
  const int* suplabel = (const int*)d_in[1];      // [16,1,1024,1024] i32
  // d_in[2..4] = seed_h(32), seed_w(32), seed_level(4): fixed by setup_inputs.
  float* out = (float*)d_out;                     // [16,3,1024,1024] f32
  float* ws = (float*)d_ws;                       // needs 16*1024*4*4B = 256KB

  sp_zero_ws<<<(NIMG * NSP + TPB - 1) / TPB, TPB, 0, stream>>>((float4*)ws);
  sp_accumulate<<<NIMG * ABLK, TPB, 0, stream>>>(input, suplabel, ws);
  sp_gather<<<NIMG * GBLK, TPB, 0, stream>>>(suplabel, ws, out);
}